// MultiViewHyperConvLayer_8864812499579
// MI455X (gfx1250) — compile-verified
//
#include <hip/hip_runtime.h>

// Two-hop COO SpMM for MI455X (gfx1250).
//   hop1: msg[u,:] += up_vals[e] * pois[up_cols[e],:]
//   hop2: out[p,:] += pu_vals[e] * msg[pu_cols[e],:]
//
// Memory/latency-bound (0.25 FLOP/byte): no matmul structure -> no WMMA.
// CDNA5 paths used:
//  - 4-deep double... quad-buffered async DMA gather per wave
//    (global_load_async_to_lds_b128 / ASYNCcnt / s_wait_asynccnt)
//  - ds_load_b128 consume from LDS
//  - hardware global_atomic_add_f32 scatter (output is L2-resident: 192MB L2
//    holds pois (102MB), msg (20.5MB) and out (102MB))

#define EMB 256
#define N_USER 20000
#define N_POI  100000
#define WAVES_PER_BLOCK 8
#define THREADS 256
#define DEPTH 4                       // async pipeline depth (ring buffers)
#define ROW_BYTES (EMB * 4)           // 1024B per embedding row

// Issue an async DMA of one 1KB embedding row (256 f32) into an LDS buffer.
// 32 lanes x 16B x 2 issues = 1024B. The immediate offset applies to BOTH the
// LDS and global addresses, so the LDS buffer exactly mirrors the row.
// ASYNCcnt += 2.
__device__ __forceinline__ void async_gather_row(unsigned lds_addr, const float* gaddr) {
  asm volatile(
      "global_load_async_to_lds_b128 %0, %1, off\n\t"
      "global_load_async_to_lds_b128 %0, %1, off offset:512"
      :
      : "v"(lds_addr), "v"(gaddr)
      : "memory");
}

#define WAIT_ASYNC(n) asm volatile("s_wait_asynccnt " #n ::: "memory")

__global__ __launch_bounds__(THREADS) void spmm_async_atomic(
    const int* __restrict__ rows, const int* __restrict__ cols,
    const float* __restrict__ vals, const float* __restrict__ dense,
    float* __restrict__ out, int nnz) {
  // per-wave ring of DEPTH 1KB buffers: 8 waves * 4KB = 32KB
  __shared__ __align__(16) float lds[WAVES_PER_BLOCK][DEPTH][EMB];

  const int lane  = threadIdx.x & 31;
  const int wave  = threadIdx.x >> 5;
  const int wstep = (blockDim.x >> 5) * gridDim.x;   // total waves in grid
  int e = blockIdx.x * (blockDim.x >> 5) + wave;     // wave-uniform edge index

  if (e >= nnz) return;  // uniform per wave

  // LDS byte offset of this wave's buffer 0, lane slot (generic ptr low 32
  // bits are the LDS offset on gfx1250)
  const unsigned lds0 =
      (unsigned)(size_t)(&lds[wave][0][0]) + (unsigned)lane * 16u;

  // prologue: fill the pipeline with up to DEPTH row-DMAs (all guards uniform)
  async_gather_row(lds0, dense + (size_t)cols[e] * EMB + lane * 4);
  if (e + 1 * wstep < nnz)
    async_gather_row(lds0 + 1 * ROW_BYTES,
                     dense + (size_t)cols[e + 1 * wstep] * EMB + lane * 4);
  if (e + 2 * wstep < nnz)
    async_gather_row(lds0 + 2 * ROW_BYTES,
                     dense + (size_t)cols[e + 2 * wstep] * EMB + lane * 4);
  if (e + 3 * wstep < nnz)
    async_gather_row(lds0 + 3 * ROW_BYTES,
                     dense + (size_t)cols[e + 3 * wstep] * EMB + lane * 4);

  int buf = 0;
  for (; e < nnz; e += wstep) {
    // Wait until only the q = #edges-in-flight-after-e (<=3) DMA pairs remain
    // outstanding -> this edge's buffer is complete. Async loads complete
    // in order, so asynccnt <= 2*q suffices. All branches wave-uniform.
    if (e + 3 * wstep < nnz)      { WAIT_ASYNC(0x6); }
    else if (e + 2 * wstep < nnz) { WAIT_ASYNC(0x4); }
    else if (e + 1 * wstep < nnz) { WAIT_ASYNC(0x2); }
    else                          { WAIT_ASYNC(0x0); }

    const float v = vals[e];
    const int   r = rows[e];

    const float4* lrow = reinterpret_cast<const float4*>(&lds[wave][buf][0]);
    const float4 a = lrow[lane];        // bytes [lane*16 .. lane*16+15]
    const float4 b = lrow[lane + 32];   // bytes [512 + lane*16 .. ]

    float* op = out + (size_t)r * EMB + lane * 4;
    // hardware global_atomic_add_f32, no return (STOREcnt path). These are
    // memory ops data-dependent on the ds_loads, ordered before the recycle
    // DMA below by the asm "memory" clobber -> no LDS overwrite hazard.
    unsafeAtomicAdd(op + 0,   a.x * v);
    unsafeAtomicAdd(op + 1,   a.y * v);
    unsafeAtomicAdd(op + 2,   a.z * v);
    unsafeAtomicAdd(op + 3,   a.w * v);
    unsafeAtomicAdd(op + 128, b.x * v);
    unsafeAtomicAdd(op + 129, b.y * v);
    unsafeAtomicAdd(op + 130, b.z * v);
    unsafeAtomicAdd(op + 131, b.w * v);

    // recycle this buffer for edge e + DEPTH*wstep
    const int en = e + DEPTH * wstep;  // uniform
    if (en < nnz)
      async_gather_row(lds0 + (unsigned)buf * ROW_BYTES,
                       dense + (size_t)cols[en] * EMB + lane * 4);

    buf = (buf + 1) & (DEPTH - 1);
  }
}

__global__ __launch_bounds__(THREADS) void zero_f4(float4* __restrict__ p, int n4) {
  int i = blockIdx.x * blockDim.x + threadIdx.x;
  int stride = gridDim.x * blockDim.x;
  float4 z; z.x = 0.f; z.y = 0.f; z.z = 0.f; z.w = 0.f;
  for (; i < n4; i += stride) p[i] = z;
}

extern "C" void kernel_launch(void* const* d_in, const int* in_sizes, int n_in,
                              void* d_out, int out_size, void* d_ws, size_t ws_size,
                              hipStream_t stream) {
  const float* pois    = (const float*)d_in[0];  // [N_POI, EMB]
  const int*   up_rows = (const int*)d_in[1];    // [NNZ] in [0, N_USER)
  const int*   up_cols = (const int*)d_in[2];    // [NNZ] in [0, N_POI)
  const float* up_vals = (const float*)d_in[3];
  const int*   pu_rows = (const int*)d_in[4];    // [NNZ] in [0, N_POI)
  const int*   pu_cols = (const int*)d_in[5];    // [NNZ] in [0, N_USER)
  const float* pu_vals = (const float*)d_in[6];

  const int nnz = in_sizes[1];

  float* out = (float*)d_out;              // [N_POI, EMB]
  float* msg = (float*)d_ws;               // [N_USER, EMB] = 20.48 MB scratch

  const dim3 blk(THREADS);
  const dim3 zgrid(2048);
  const dim3 sgrid(2048);                  // 16384 waves, ~18 edges each

  // zero accumulators (d_out / d_ws are poisoned by the harness)
  zero_f4<<<zgrid, blk, 0, stream>>>((float4*)msg, N_USER * EMB / 4);
  zero_f4<<<zgrid, blk, 0, stream>>>((float4*)out, N_POI * EMB / 4);

  // hop 1: msg = HG_up @ pois
  spmm_async_atomic<<<sgrid, blk, 0, stream>>>(up_rows, up_cols, up_vals,
                                               pois, msg, nnz);
  // hop 2: out = HG_pu @ msg
  spmm_async_atomic<<<sgrid, blk, 0, stream>>>(pu_rows, pu_cols, pu_vals,
                                               msg, out, nnz);
}